// transform_network_61125974556932
// MI455X (gfx1250) — compile-verified
//
#include <hip/hip_runtime.h>

// MI455X / gfx1250, wave32. FP32 WMMA (V_WMMA_F32_16X16X4_F32) keeps the
// reference's fp32 precision while moving the 5x5-conv dot products onto the
// matrix pipe. LDS staging for input tiles + filters; branch-free lane masks
// so EXEC is all-ones at every WMMA.

typedef __attribute__((ext_vector_type(2))) float v2f;
typedef __attribute__((ext_vector_type(8))) float v8f;

// ---------------------------------------------------------------------------
// Forward: 5x5 conv, stride 2, pad 2, 1 -> 8 channels.
// Block = 256 threads (8 waves), computes a 16x16 output tile (all 8 ch).
// Per wave: 2 output rows; per row: M=16 out pixels, N=16 (8 valid ch),
// K=25 padded to 28 -> 7 WMMA f32 16x16x4 steps.
// ---------------------------------------------------------------------------
__global__ __launch_bounds__(256) void fwd_conv5x5_s2_wmma(
    const float* __restrict__ in, long inBatchStride,
    const float* __restrict__ filt,      // 8*25 fp32
    float* __restrict__ out,             // (32,8,Ho,Wo)
    int H, int W, int Ho, int Wo)
{
  const int col0 = blockIdx.x * 16;
  const int row0 = blockIdx.y * 16;
  const int b    = blockIdx.z;
  const int tid  = threadIdx.x;
  const int wv   = tid >> 5;
  const int lane = tid & 31;
  const int hf   = lane >> 4;      // which lane-half (K pairing)
  const int nl   = lane & 15;      // M index (A) / N index (B,D)

  __shared__ float sIn[36 * 36];   // input tile + halo, zero-padded
  __shared__ float sF[200];        // 8 filters * 25 taps

  if (tid == 0) __builtin_prefetch(filt, 0, 1);
  for (int i = tid; i < 200; i += 256) sF[i] = filt[i];

  const int iy0 = 2 * row0 - 2;
  const int ix0 = 2 * col0 - 2;
  for (int i = tid; i < 36 * 36; i += 256) {
    int r = i / 36, c = i - r * 36;
    int gy = iy0 + r, gx = ix0 + c;
    float v = 0.0f;
    if ((unsigned)gy < (unsigned)H && (unsigned)gx < (unsigned)W)
      v = in[(long)b * inBatchStride + (long)gy * W + gx];
    sIn[i] = v;
  }
  __syncthreads();

  // B fragments: weight[orient=nl][k], k = 4s + 2*hf + {0,1}; zero-pad k>=25
  // and orientations 8..15. Constant across the row loop -> precompute.
  v2f Bv[7];
#pragma unroll
  for (int s = 0; s < 7; ++s) {
    int k0 = 4 * s + 2 * hf;
    Bv[s][0] = (nl < 8 && k0     < 25) ? sF[nl * 25 + k0]     : 0.0f;
    Bv[s][1] = (nl < 8 && k0 + 1 < 25) ? sF[nl * 25 + k0 + 1] : 0.0f;
  }

#pragma unroll
  for (int rr = 0; rr < 2; ++rr) {
    const int r = wv * 2 + rr;     // local output row 0..15
    v8f acc = {};
#pragma unroll
    for (int s = 0; s < 7; ++s) {
      // A fragment: patch[pixel m=nl][k]. For padded k (>=25) the kh/kw
      // address stays inside the 36x36 tile and B is zero, so no mask needed.
      int k0 = 4 * s + 2 * hf;
      int kh0 = k0 / 5,       kw0 = k0 - kh0 * 5;
      int k1  = k0 + 1;
      int kh1 = k1 / 5,       kw1 = k1 - kh1 * 5;
      v2f Av;
      Av[0] = sIn[(2 * r + kh0) * 36 + (2 * nl + kw0)];
      Av[1] = sIn[(2 * r + kh1) * 36 + (2 * nl + kw1)];
      acc = __builtin_amdgcn_wmma_f32_16x16x4_f32(
          false, Av, false, Bv[s], (short)0, acc, false, false);
    }
    // D layout: VGPR j -> M = j + 8*hf, N = nl. Lanes nl<8 hold real channels.
    if (nl < 8) {
      const int gy = row0 + r;
#pragma unroll
      for (int j = 0; j < 8; ++j) {
        int gx = col0 + j + 8 * hf;
        out[(((long)b * 8 + nl) * Ho + gy) * Wo + gx] = acc[j];
      }
    }
  }
}

// ---------------------------------------------------------------------------
// Inverse: 5x5 transposed conv, stride 2 (pad ip=2, out_pad 1), 8 -> 1 ch.
// Polyphase: out[2by+py][2bx+px] = sum_{c,dy,dx} in[c][by+dy][bx+dx] *
//   inv_f[c][2+py-2dy][2+px-2dx]   (taps with index outside [0,4] are zero).
// Block = 256 threads, base tile 8 rows x 16 cols -> 32x16 output pixels.
// Per wave: one base row; M=16 base cols, N=16 (4 valid phases), K=72
// -> 18 WMMA f32 16x16x4 steps.
// ---------------------------------------------------------------------------
__global__ __launch_bounds__(256) void inv_tconv5x5_s2_wmma(
    const float* __restrict__ rec,  long recBatchStride, // channel-0 plane (H*W)
    const float* __restrict__ coef,                      // (32,8,H,W), use ch 1..7
    const float* __restrict__ filt,                      // inv_f 8*25 fp32
    float* __restrict__ outp,                            // (32,1,2H,2W)
    int H, int W)
{
  const int bx0 = blockIdx.x * 16;
  const int by0 = blockIdx.y * 8;
  const int b   = blockIdx.z;
  const int tid = threadIdx.x;
  const int wv  = tid >> 5;
  const int lane = tid & 31;
  const int hf  = lane >> 4;
  const int nl  = lane & 15;

  __shared__ float sIn[8 * 10 * 18]; // 8 ch x (8 rows + halo) x (16 cols + halo)
  __shared__ float sF[200];

  for (int i = tid; i < 200; i += 256) sF[i] = filt[i];

  for (int i = tid; i < 8 * 10 * 18; i += 256) {
    int c = i / 180, rem = i - c * 180;
    int r = rem / 18, cc = rem - r * 18;
    int gy = by0 + r - 1, gx = bx0 + cc - 1;
    float v = 0.0f;
    if ((unsigned)gy < (unsigned)H && (unsigned)gx < (unsigned)W) {
      v = (c == 0) ? rec[(long)b * recBatchStride + (long)gy * W + gx]
                   : coef[(((long)b * 8 + c) * H + gy) * W + gx];
    }
    sIn[i] = v;
  }
  __syncthreads();

  const int py = (nl & 3) >> 1;   // phase row (valid for lanes nl<4)
  const int px = nl & 1;          // phase col
  const int r  = wv;              // local base row 0..7

  v8f acc = {};
#pragma unroll
  for (int s = 0; s < 18; ++s) {
    int k0 = 4 * s + 2 * hf;
    v2f Av, Bv;
#pragma unroll
    for (int t = 0; t < 2; ++t) {
      int k  = k0 + t;               // 0..71, K=72 exactly
      int c  = k / 9;
      int r9 = k - c * 9;
      int dy = r9 / 3 - 1;
      int dx = r9 - (r9 / 3) * 3 - 1;
      // A: input patch for base col m=nl
      float a = sIn[c * 180 + (r + dy + 1) * 18 + (nl + dx + 1)];
      // B: phase weight; ih/iw in {0..5}, 5 => tap outside kernel => 0
      int ih = 2 + py - 2 * dy;
      int iw = 2 + px - 2 * dx;
      float w = (nl < 4 && ih < 5 && iw < 5) ? sF[c * 25 + ih * 5 + iw] : 0.0f;
      Av[t] = a;
      Bv[t] = w;
    }
    acc = __builtin_amdgcn_wmma_f32_16x16x4_f32(
        false, Av, false, Bv, (short)0, acc, false, false);
  }

  // D: VGPR j -> base col m = j + 8*hf, N = nl (phase). Lanes nl<4 store.
  if (nl < 4) {
    const int oy = 2 * (by0 + r) + py;
    const long oplane = (long)b * (2 * H);
#pragma unroll
    for (int j = 0; j < 8; ++j) {
      int ox = 2 * (bx0 + j + 8 * hf) + px;
      outp[(oplane + oy) * (2L * W) + ox] = acc[j];
    }
  }
}

// ---------------------------------------------------------------------------
// Orchestration: 4 forward levels then 4 inverse levels, all on `stream`.
// ---------------------------------------------------------------------------
extern "C" void kernel_launch(void* const* d_in, const int* in_sizes, int n_in,
                              void* d_out, int out_size, void* d_ws, size_t ws_size,
                              hipStream_t stream) {
  (void)in_sizes; (void)n_in; (void)out_size; (void)ws_size;
  const float* x     = (const float*)d_in[0];   // (32,1,512,512)
  const float* fwd_f = (const float*)d_in[1];   // (8,1,5,5)
  const float* inv_f = (const float*)d_in[2];   // (8,1,5,5)
  float* out = (float*)d_out;                   // (32,1,512,512)
  float* ws  = (float*)d_ws;

  // workspace layout (floats)
  const size_t n4 = 32UL * 8 * 256 * 256;
  const size_t n3 = 32UL * 8 * 128 * 128;
  const size_t n2 = 32UL * 8 * 64 * 64;
  const size_t n1 = 32UL * 8 * 32 * 32;
  float* res4 = ws;
  float* res3 = res4 + n4;
  float* res2 = res3 + n3;
  float* res1 = res2 + n2;
  float* rec1 = res1 + n1;                      // 32*64*64
  float* rec2 = rec1 + 32UL * 64 * 64;          // 32*128*128
  float* rec3 = rec2 + 32UL * 128 * 128;        // 32*256*256

  const dim3 blk(256);

  // Forward pyramid (input of levels 3..1 is channel 0 of the previous level)
  fwd_conv5x5_s2_wmma<<<dim3(16, 16, 32), blk, 0, stream>>>(
      x,    512L * 512,      fwd_f, res4, 512, 512, 256, 256);
  fwd_conv5x5_s2_wmma<<<dim3(8, 8, 32), blk, 0, stream>>>(
      res4, 8L * 256 * 256,  fwd_f, res3, 256, 256, 128, 128);
  fwd_conv5x5_s2_wmma<<<dim3(4, 4, 32), blk, 0, stream>>>(
      res3, 8L * 128 * 128,  fwd_f, res2, 128, 128, 64, 64);
  fwd_conv5x5_s2_wmma<<<dim3(2, 2, 32), blk, 0, stream>>>(
      res2, 8L * 64 * 64,    fwd_f, res1, 64, 64, 32, 32);

  // Inverse pyramid. Level 1: coeffs[0] == res1 channel 0, so rec = res1 ch0.
  inv_tconv5x5_s2_wmma<<<dim3(2, 4, 32), blk, 0, stream>>>(
      res1, 8L * 32 * 32,    res1, inv_f, rec1, 32, 32);
  inv_tconv5x5_s2_wmma<<<dim3(4, 8, 32), blk, 0, stream>>>(
      rec1, 64L * 64,        res2, inv_f, rec2, 64, 64);
  inv_tconv5x5_s2_wmma<<<dim3(8, 16, 32), blk, 0, stream>>>(
      rec2, 128L * 128,      res3, inv_f, rec3, 128, 128);
  inv_tconv5x5_s2_wmma<<<dim3(16, 32, 32), blk, 0, stream>>>(
      rec3, 256L * 256,      res4, inv_f, out, 256, 256);
}